// Induction_1726576856433
// MI455X (gfx1250) — compile-verified
//
#include <hip/hip_runtime.h>

typedef __attribute__((ext_vector_type(2))) float v2f;
typedef __attribute__((ext_vector_type(8))) float v8f;

#define C_CAPS 32
#define S_DIM  2048
#define H_DIM  1024
#define NCHUNK 8
#define SCHUNK (S_DIM / NCHUNK)   // 256

// ---------------------------------------------------------------------------
// Pass A: xd[c][h] = sum_s d[c][s] * x[c][s][h]   (d == nullptr -> uniform 1/S)
// Grid: (NCHUNK, C_CAPS), 256 threads. Each block streams SCHUNK rows of x,
// full H covered by 256 threads * float4. Partials out per chunk.
// ---------------------------------------------------------------------------
__global__ __launch_bounds__(256) void k_weighted_sum(
    const float* __restrict__ x, const float* __restrict__ d,
    float* __restrict__ partials)
{
    __shared__ float wts[SCHUNK];
    const int c     = blockIdx.y;
    const int chunk = blockIdx.x;
    const int t     = threadIdx.x;
    const int s0    = chunk * SCHUNK;

    if (d) wts[t] = d[c * S_DIM + s0 + t];
    else   wts[t] = 1.0f / (float)S_DIM;
    __syncthreads();

    const float4* xrow =
        reinterpret_cast<const float4*>(x + (size_t)c * S_DIM * H_DIM +
                                        (size_t)s0 * H_DIM) + t;
    float4 acc = make_float4(0.f, 0.f, 0.f, 0.f);
#pragma unroll 4
    for (int s = 0; s < SCHUNK; ++s) {
        float4 v = xrow[s * (H_DIM / 4)];
        float  w = wts[s];
        acc.x += w * v.x; acc.y += w * v.y;
        acc.z += w * v.z; acc.w += w * v.w;
    }
    float4* p = reinterpret_cast<float4*>(
        partials + ((size_t)chunk * C_CAPS + c) * H_DIM) + t;
    *p = acc;
}

// Reduce the NCHUNK partials into xd (C*H elements). Grid: C*H/256 blocks.
__global__ __launch_bounds__(256) void k_reduce_partials(
    const float* __restrict__ partials, float* __restrict__ xd)
{
    const int i = blockIdx.x * 256 + threadIdx.x;
    float s = 0.f;
#pragma unroll
    for (int k = 0; k < NCHUNK; ++k)
        s += partials[(size_t)k * C_CAPS * H_DIM + i];
    xd[i] = s;
}

// ---------------------------------------------------------------------------
// WMMA fp32 GEMM: Dst(32 x H) = A(32 x H) * B(H x H)
//   BT=false: B[k][n] = W[k*H + n]      (c_raw = xd @ W)
//   BT=true : B[k][n] = W[n*H + k]      (y     = c  @ W^T)
// One wave per block computes a 32x16 tile of Dst (2 M-tiles sharing B).
// v_wmma_f32_16x16x4_f32 ISA layouts (ISA 7.12.2):
//   A 16x4: lanes 0-15 hold K=0,1 ; lanes 16-31 hold K=2,3 (row = lane&15)
//   D 16x16: VGPR i -> row i (lanes<16) / row i+8 (lanes>=16), col = lane&15
// ---------------------------------------------------------------------------
template <bool BT>
__global__ __launch_bounds__(32) void k_gemm32(
    const float* __restrict__ A, const float* __restrict__ W,
    float* __restrict__ Dst)
{
    const int lane = threadIdx.x;
    const int n0   = blockIdx.x * 16;
    const int kq   = (lane >> 4) * 2;   // 0 or 2
    const int l    = lane & 15;

    v8f c0 = {}; v8f c1 = {};
    for (int k = 0; k < H_DIM; k += 4) {
        const int kk = k + kq;
        v2f a0 = *(const v2f*)(A + (size_t)l * H_DIM + kk);
        v2f a1 = *(const v2f*)(A + (size_t)(l + 16) * H_DIM + kk);
        v2f b;
        if (BT) {
            b = *(const v2f*)(W + (size_t)(n0 + l) * H_DIM + kk);
        } else {
            b.x = W[(size_t)kk * H_DIM + n0 + l];
            b.y = W[(size_t)(kk + 1) * H_DIM + n0 + l];
        }
        c0 = __builtin_amdgcn_wmma_f32_16x16x4_f32(
                 false, a0, false, b, (short)0, c0, false, false);
        c1 = __builtin_amdgcn_wmma_f32_16x16x4_f32(
                 false, a1, false, b, (short)0, c1, false, false);
    }
    const int rbase = (lane >> 4) * 8;
    const int col   = n0 + l;
#pragma unroll
    for (int i = 0; i < 8; ++i) {
        Dst[(size_t)(rbase + i) * H_DIM + col]      = c0[i];
        Dst[(size_t)(rbase + i + 16) * H_DIM + col] = c1[i];
    }
}

// ---------------------------------------------------------------------------
// Squash: out[c] = coeff(||craw[c]||^2) * craw[c]. One block per capsule.
// ---------------------------------------------------------------------------
__global__ __launch_bounds__(256) void k_squash(
    const float* __restrict__ craw, float* __restrict__ out)
{
    __shared__ float red[256];
    const int c = blockIdx.x;
    const int t = threadIdx.x;
    float4 v = reinterpret_cast<const float4*>(craw + (size_t)c * H_DIM)[t];
    red[t] = v.x * v.x + v.y * v.y + v.z * v.z + v.w * v.w;
    __syncthreads();
    for (int off = 128; off > 0; off >>= 1) {
        if (t < off) red[t] += red[t + off];
        __syncthreads();
    }
    const float sq    = red[0];
    const float coeff = sq / ((1.0f + sq) * sqrtf(sq + 1e-9f));
    float4 r = make_float4(coeff * v.x, coeff * v.y, coeff * v.z, coeff * v.w);
    reinterpret_cast<float4*>(out + (size_t)c * H_DIM)[t] = r;
}

// ---------------------------------------------------------------------------
// Pass C: b[c][s] (+)= dot(x[c][s][:], y[c][:]). One wave per (c,s) row,
// 8 waves per block, wave32 shuffle reduction. Grid: C*S/8 blocks.
// ---------------------------------------------------------------------------
__global__ __launch_bounds__(256) void k_dot_update(
    const float* __restrict__ x, const float* __restrict__ y,
    float* __restrict__ b, int zero_b)
{
    const int wave = threadIdx.x >> 5;
    const int lane = threadIdx.x & 31;
    const int row  = blockIdx.x * 8 + wave;     // row in [0, C*S)
    const int c    = row >> 11;                 // row / S_DIM

    const float4* xr = reinterpret_cast<const float4*>(x + (size_t)row * H_DIM);
    const float4* yr = reinterpret_cast<const float4*>(y + (size_t)c * H_DIM);
    float acc = 0.f;
#pragma unroll
    for (int j = 0; j < 8; ++j) {
        float4 xv = xr[j * 32 + lane];
        float4 yv = yr[j * 32 + lane];
        acc += xv.x * yv.x + xv.y * yv.y + xv.z * yv.z + xv.w * yv.w;
    }
#pragma unroll
    for (int off = 16; off > 0; off >>= 1)
        acc += __shfl_down(acc, off, 32);
    if (lane == 0) {
        float prev = zero_b ? 0.f : b[row];
        b[row] = prev + acc;
    }
}

// ---------------------------------------------------------------------------
// Softmax over S per capsule. One block per capsule, 8 elems per thread.
// ---------------------------------------------------------------------------
__global__ __launch_bounds__(256) void k_softmax(
    const float* __restrict__ b, float* __restrict__ d)
{
    __shared__ float red[256];
    const int c = blockIdx.x;
    const int t = threadIdx.x;
    const float* row = b + (size_t)c * S_DIM;

    float vals[8];
    float m = -__builtin_inff();
#pragma unroll
    for (int j = 0; j < 8; ++j) {
        vals[j] = row[j * 256 + t];
        m = fmaxf(m, vals[j]);
    }
    red[t] = m; __syncthreads();
    for (int off = 128; off > 0; off >>= 1) {
        if (t < off) red[t] = fmaxf(red[t], red[t + off]);
        __syncthreads();
    }
    m = red[0];
    __syncthreads();

    float sum = 0.f;
#pragma unroll
    for (int j = 0; j < 8; ++j) {
        vals[j] = __expf(vals[j] - m);
        sum += vals[j];
    }
    red[t] = sum; __syncthreads();
    for (int off = 128; off > 0; off >>= 1) {
        if (t < off) red[t] += red[t + off];
        __syncthreads();
    }
    const float inv = 1.0f / red[0];
    float* drow = d + (size_t)c * S_DIM;
#pragma unroll
    for (int j = 0; j < 8; ++j) drow[j * 256 + t] = vals[j] * inv;
}

// ---------------------------------------------------------------------------
extern "C" void kernel_launch(void* const* d_in, const int* in_sizes, int n_in,
                              void* d_out, int out_size, void* d_ws, size_t ws_size,
                              hipStream_t stream)
{
    const float* x = (const float*)d_in[0];   // (C,S,H) fp32
    const float* W = (const float*)d_in[1];   // (H,H)   fp32
    float* out = (float*)d_out;               // (C,H)   fp32
    float* ws  = (float*)d_ws;

    float* partials = ws;                                   // NCHUNK*C*H
    float* xd   = partials + (size_t)NCHUNK * C_CAPS * H_DIM;
    float* craw = xd   + C_CAPS * H_DIM;
    float* cvec = craw + C_CAPS * H_DIM;
    float* yv   = cvec + C_CAPS * H_DIM;
    float* bl   = yv   + C_CAPS * H_DIM;                    // C*S
    float* dw   = bl   + C_CAPS * S_DIM;                    // C*S

    const dim3 gws(NCHUNK, C_CAPS);
    const int  gred = (C_CAPS * H_DIM) / 256;               // 128
    const int  gnt  = H_DIM / 16;                           // 64
    const int  gdot = (C_CAPS * S_DIM) / 8;                 // 8192

    // ---- iteration 1 (b = 0 -> uniform d) ----
    k_weighted_sum<<<gws, 256, 0, stream>>>(x, nullptr, partials);
    k_reduce_partials<<<gred, 256, 0, stream>>>(partials, xd);
    k_gemm32<false><<<gnt, 32, 0, stream>>>(xd, W, craw);
    k_squash<<<C_CAPS, 256, 0, stream>>>(craw, cvec);
    k_gemm32<true><<<gnt, 32, 0, stream>>>(cvec, W, yv);
    k_dot_update<<<gdot, 256, 0, stream>>>(x, yv, bl, 1);

    // ---- iteration 2 ----
    k_softmax<<<C_CAPS, 256, 0, stream>>>(bl, dw);
    k_weighted_sum<<<gws, 256, 0, stream>>>(x, dw, partials);
    k_reduce_partials<<<gred, 256, 0, stream>>>(partials, xd);
    k_gemm32<false><<<gnt, 32, 0, stream>>>(xd, W, craw);
    k_squash<<<C_CAPS, 256, 0, stream>>>(craw, cvec);
    k_gemm32<true><<<gnt, 32, 0, stream>>>(cvec, W, yv);
    k_dot_update<<<gdot, 256, 0, stream>>>(x, yv, bl, 0);

    // ---- iteration 3 (final c only; b update is dead code) ----
    k_softmax<<<C_CAPS, 256, 0, stream>>>(bl, dw);
    k_weighted_sum<<<gws, 256, 0, stream>>>(x, dw, partials);
    k_reduce_partials<<<gred, 256, 0, stream>>>(partials, xd);
    k_gemm32<false><<<gnt, 32, 0, stream>>>(xd, W, craw);
    k_squash<<<C_CAPS, 256, 0, stream>>>(craw, out);
}